// SGL_5884105195912
// MI455X (gfx1250) — compile-verified
//
#include <hip/hip_runtime.h>
#include <hip/hip_bf16.h>

// LightGCN propagation: 3 x (COO SpMM + accumulate), mean over 4 embeddings.
// Memory-bound (AI ~0.25 flop/byte). MI455X strategy:
//   - cur/nxt ping-pong buffers (153.6 MB) kept L2-resident (192 MB L2):
//     all gathers + fp32 atomics resolve in L2.
//   - everything read/written exactly once (edge arrays, acc/d_out, inputs)
//     uses non-temporal hints so it cannot evict the resident buffers.
//   - per-block async staging of edge triples into LDS
//     (global_load_async_to_lds_b32 + s_wait_asynccnt).
//   - 16 lanes/edge: one global_load_b128 per lane covers a 256B row per
//     half-wave32; 4 consecutive global_atomic_add_f32 per lane for scatter.

#define NUM_USERS 200000
#define NUM_ITEMS 100000
#define HIDDEN    64
#define N_NODES   (NUM_USERS + NUM_ITEMS)
#define N_LAYERS  3
#define V4_PER_NODE (HIDDEN / 4)                      // 16 float4 per node row
#define TOTAL_V4  ((long)N_NODES * V4_PER_NODE)       // 4.8M float4
#define EDGES_PER_BLOCK 256

typedef float f4 __attribute__((ext_vector_type(4)));

#if defined(__has_builtin)
#if __has_builtin(__builtin_amdgcn_global_load_async_to_lds_b32)
#define HAVE_ASYNC_LDS 1
#endif
#endif

#if defined(HAVE_ASYNC_LDS)
typedef __attribute__((address_space(1))) int g_i32;  // global int (per clang sig)
typedef __attribute__((address_space(3))) int l_i32;  // LDS int
#endif

__global__ __launch_bounds__(256) void lgcn_init(
    const f4* __restrict__ user_emb,
    const f4* __restrict__ item_emb,
    f4* __restrict__ acc,      // d_out, running sum (streaming -> NT)
    f4* __restrict__ cur,      // current layer input (keep in L2 -> RT)
    f4* __restrict__ nxt)      // next layer accumulator (keep in L2 -> RT)
{
    long i = (long)blockIdx.x * blockDim.x + threadIdx.x;
    if (i >= TOTAL_V4) return;
    const long userV4 = (long)NUM_USERS * V4_PER_NODE;
    const f4* src = (i < userV4) ? (user_emb + i) : (item_emb + (i - userV4));
    f4 e = __builtin_nontemporal_load(src);           // read-once stream
    __builtin_nontemporal_store(e, acc + i);          // streaming accumulator
    cur[i] = e;                                       // RT: gather target
    f4 z = {0.f, 0.f, 0.f, 0.f};
    nxt[i] = z;                                       // RT: atomic target
}

// Block owns EDGES_PER_BLOCK edges: stage (row,col,val) triples into LDS
// (async on gfx1250), then 16 passes where each 16-lane group processes one
// edge: b128 gather from cur[col] (L2-resident), scale, 4x f32 atomics into
// nxt[row] (L2-resident).
__global__ __launch_bounds__(256) void lgcn_spmm(
    const int*   __restrict__ rows,
    const int*   __restrict__ cols,
    const float* __restrict__ vals,
    const f4*    __restrict__ cur,
    float*       __restrict__ nxt,
    int n_edges)
{
    __shared__ int   s_rows[EDGES_PER_BLOCK];
    __shared__ int   s_cols[EDGES_PER_BLOCK];
    __shared__ float s_vals[EDGES_PER_BLOCK];

    const int tid  = threadIdx.x;
    const int base = blockIdx.x * EDGES_PER_BLOCK;

    // Stream-ahead prefetch of edge arrays (gfx1250 global_prefetch_b8).
    if (base + tid + 8192 < n_edges) {
        __builtin_prefetch(rows + base + tid + 8192, 0, 0);
        __builtin_prefetch(cols + base + tid + 8192, 0, 0);
        __builtin_prefetch(vals + base + tid + 8192, 0, 0);
    }

    if (base + tid < n_edges) {
#if defined(HAVE_ASYNC_LDS)
        __builtin_amdgcn_global_load_async_to_lds_b32(
            (g_i32*)(void*)(rows + base + tid),
            (l_i32*)(void*)(s_rows + tid), 0, 0);
        __builtin_amdgcn_global_load_async_to_lds_b32(
            (g_i32*)(void*)(cols + base + tid),
            (l_i32*)(void*)(s_cols + tid), 0, 0);
        __builtin_amdgcn_global_load_async_to_lds_b32(
            (g_i32*)(void*)(vals + base + tid),
            (l_i32*)(void*)(s_vals + tid), 0, 0);
#else
        s_rows[tid] = __builtin_nontemporal_load(rows + base + tid);
        s_cols[tid] = __builtin_nontemporal_load(cols + base + tid);
        s_vals[tid] = __builtin_nontemporal_load(vals + base + tid);
#endif
    }
#if defined(HAVE_ASYNC_LDS)
#if __has_builtin(__builtin_amdgcn_s_wait_asynccnt)
    __builtin_amdgcn_s_wait_asynccnt(0);
#else
    asm volatile("s_wait_asynccnt 0x0" ::: "memory");
#endif
#endif
    __syncthreads();

    const int slot = tid >> 4;                        // 16 edge slots per pass
    const int sub  = tid & 15;                        // float4 lane within row
    #pragma unroll 4
    for (int it = 0; it < EDGES_PER_BLOCK / 16; ++it) {
        const int le = it * 16 + slot;                // local edge 0..255
        const int e  = base + le;
        if (e < n_edges) {
            int   r = s_rows[le];                     // ds_load broadcast
            int   c = s_cols[le];
            float v = s_vals[le];

            f4 x = cur[(long)c * V4_PER_NODE + sub];  // global_load_b128, RT
            float* dst = nxt + (long)r * HIDDEN + sub * 4;

            unsafeAtomicAdd(dst + 0, x.x * v);        // global_atomic_add_f32
            unsafeAtomicAdd(dst + 1, x.y * v);
            unsafeAtomicAdd(dst + 2, x.z * v);
            unsafeAtomicAdd(dst + 3, x.w * v);
        }
    }
}

// acc += nxt; cur = nxt; nxt = 0.  Last layer: acc = (acc+nxt)/4 -> d_out.
__global__ __launch_bounds__(256) void lgcn_fold(
    f4* __restrict__ acc,
    f4* __restrict__ cur,
    f4* __restrict__ nxt,
    int is_last)
{
    long i = (long)blockIdx.x * blockDim.x + threadIdx.x;
    if (i >= TOTAL_V4) return;
    f4 v = nxt[i];                                    // RT (stays resident)
    f4 a = __builtin_nontemporal_load(acc + i);       // streaming accumulator
    a = a + v;
    if (is_last) {
        a = a * 0.25f;
        __builtin_nontemporal_store(a, acc + i);
    } else {
        __builtin_nontemporal_store(a, acc + i);
        cur[i] = v;                                   // RT: next gather target
        f4 z = {0.f, 0.f, 0.f, 0.f};
        nxt[i] = z;                                   // RT: next atomic target
    }
}

extern "C" void kernel_launch(void* const* d_in, const int* in_sizes, int n_in,
                              void* d_out, int out_size, void* d_ws, size_t ws_size,
                              hipStream_t stream) {
    const int*   adj_row  = (const int*)  d_in[0];
    const int*   adj_col  = (const int*)  d_in[1];
    const float* adj_vals = (const float*)d_in[2];
    const f4*    user_emb = (const f4*)   d_in[3];
    const f4*    item_emb = (const f4*)   d_in[4];
    const int n_edges = in_sizes[0];

    float* acc = (float*)d_out;                       // accumulator lives in d_out
    float* cur = (float*)d_ws;                        // 76.8 MB
    float* nxt = cur + (size_t)N_NODES * HIDDEN;      // 76.8 MB (ws >= 153.6 MB)

    const int B = 256;
    const int nodeBlocks = (int)((TOTAL_V4 + B - 1) / B);

    lgcn_init<<<nodeBlocks, B, 0, stream>>>(user_emb, item_emb,
                                            (f4*)acc, (f4*)cur, (f4*)nxt);

    const int spmmBlocks = (n_edges + EDGES_PER_BLOCK - 1) / EDGES_PER_BLOCK;

    for (int l = 0; l < N_LAYERS; ++l) {
        lgcn_spmm<<<spmmBlocks, B, 0, stream>>>(adj_row, adj_col, adj_vals,
                                                (const f4*)cur, nxt, n_edges);
        lgcn_fold<<<nodeBlocks, B, 0, stream>>>((f4*)acc, (f4*)cur, (f4*)nxt,
                                                l == N_LAYERS - 1 ? 1 : 0);
    }
}